// MambaController_14620068675847
// MI455X (gfx1250) — compile-verified
//
#include <hip/hip_runtime.h>

typedef float v2f __attribute__((ext_vector_type(2)));
typedef float v8f __attribute__((ext_vector_type(8)));
typedef int   v4i __attribute__((ext_vector_type(4)));

#define T_TOK   4096      // B*L
#define BB      4
#define LL      1024
#define DMODEL  1024
#define DINNER  2048
#define NSTATE  16
#define RRANK   64
#define XDBL    96        // R + 2*NST
#define NLAY    2
#define OBSD    128
#define ACTD    32

#if __has_builtin(__builtin_amdgcn_global_load_async_to_lds_b128) && \
    __has_builtin(__builtin_amdgcn_s_wait_asynccnt)
#define ASYNC_LDS 1
typedef __attribute__((address_space(1))) v4i* gv4i_p;   // global
typedef __attribute__((address_space(3))) v4i* lv4i_p;   // LDS
#else
#define ASYNC_LDS 0
#endif

// 16-byte global -> LDS copy (async on CDNA5, sync fallback otherwise)
__device__ __forceinline__ void cp_b128(float* lds, const float* g) {
#if ASYNC_LDS
  __builtin_amdgcn_global_load_async_to_lds_b128((gv4i_p)g, (lv4i_p)lds, 0, 0);
#else
  *(float4*)lds = *(const float4*)g;
#endif
}
template <int N>
__device__ __forceinline__ void cp_wait() {
#if ASYNC_LDS
  __builtin_amdgcn_s_wait_asynccnt(N);
#endif
}

// ---------------------------------------------------------------------------
// C[M,N] = A[M,K] @ W[N,K]^T (+bias) (+activation)   via V_WMMA_F32_16X16X4_F32
// block: 256 thr = 8 waves in 4(M)x2(N) grid; block tile 128(M) x 128(N);
// each wave owns 32(M) x 64(N) = 8 f32 accumulators.
// K staged in 16-wide chunks through double-buffered LDS with async copies.
// act: 0 = none, 1 = softplus.  M == 4096 (gridDim.y fixed), K % 16 == 0.
// ---------------------------------------------------------------------------
__global__ __launch_bounds__(256) void gemm_wmma_f32(
    const float* __restrict__ A, int lda,
    const float* __restrict__ W, int ldw,
    const float* __restrict__ bias,
    float* __restrict__ C, int ldc,
    int N, int K, int act)
{
  __shared__ float sA[2][128 * 20];   // stride 20: 16B aligned, conflict-free
  __shared__ float sB[2][128 * 20];

  const int tid  = threadIdx.x;
  const int wave = tid >> 5;
  const int lane = tid & 31;
  const int hi   = lane >> 4;         // 0: K={k,k+1}, 1: K={k+2,k+3}
  const int lr   = lane & 15;
  const int m0   = (wave >> 1) * 32;  // wave's M offset in block tile
  const int n0   = (wave & 1) * 64;   // wave's N offset in block tile

  const int mBlk = blockIdx.y * 128;
  const int nBlk = blockIdx.x * 128;

  // stage one 16-wide K chunk into LDS buffer bb (exactly 4 copies/thread)
  auto stage = [&](int bb, int k0) {
#pragma unroll
    for (int it = 0; it < 2; ++it) {
      int i   = tid + it * 256;
      int row = i >> 2;
      int c4  = (i & 3) * 4;
      cp_b128(&sA[bb][row * 20 + c4], A + (size_t)(mBlk + row) * lda + k0 + c4);
      int wr  = nBlk + row;
      if (wr >= N) wr = N - 1;        // clamp: garbage only feeds masked cols
      cp_b128(&sB[bb][row * 20 + c4], W + (size_t)wr * ldw + k0 + c4);
    }
  };

  v8f accA[4] = {}, accB[4] = {};

  const int nch = K >> 4;
  stage(0, 0);
  for (int c = 0; c < nch; ++c) {
    const int cur = c & 1;
    if (c + 1 < nch) { stage(1 - cur, (c + 1) * 16); cp_wait<4>(); }
    else             { cp_wait<0>(); }
    __syncthreads();

    const float* pA = sA[cur];
    const float* pB = sB[cur];
#pragma unroll
    for (int k4 = 0; k4 < 16; k4 += 4) {
      v2f a0 = *(const v2f*)(pA + (m0 +      lr) * 20 + k4 + 2 * hi);
      v2f a1 = *(const v2f*)(pA + (m0 + 16 + lr) * 20 + k4 + 2 * hi);
#pragma unroll
      for (int j = 0; j < 4; ++j) {
        v2f b = *(const v2f*)(pB + (n0 + j * 16 + lr) * 20 + k4 + 2 * hi);
        accA[j] = __builtin_amdgcn_wmma_f32_16x16x4_f32(false, a0, false, b, (short)0, accA[j], false, false);
        accB[j] = __builtin_amdgcn_wmma_f32_16x16x4_f32(false, a1, false, b, (short)0, accB[j], false, false);
      }
    }
    __syncthreads();
  }

  // epilogue: C layout per ISA: VGPR r -> row m + r + 8*hi, col n + lr
  const int mBaseA = mBlk + m0 + hi * 8;
  const int mBaseB = mBaseA + 16;
#pragma unroll
  for (int j = 0; j < 4; ++j) {
    int col = nBlk + n0 + j * 16 + lr;
    if (col < N) {
      float bv = bias ? bias[col] : 0.f;
#pragma unroll
      for (int r = 0; r < 8; ++r) {
        float vA = accA[j][r] + bv;
        float vB = accB[j][r] + bv;
        if (act == 1) {
          vA = (vA > 20.f) ? vA : logf(1.f + __expf(vA));   // softplus
          vB = (vB > 20.f) ? vB : logf(1.f + __expf(vB));
        }
        C[(size_t)(mBaseA + r) * ldc + col] = vA;
        C[(size_t)(mBaseB + r) * ldc + col] = vB;
      }
    }
  }
}

// ---------------------------------------------------------------------------
// depthwise causal conv (K=4) + bias + SiLU.  xz: (T, 4096), xc = cols 0..2047
// ---------------------------------------------------------------------------
__global__ __launch_bounds__(256) void conv_silu(
    const float* __restrict__ xz, const float* __restrict__ cw,
    const float* __restrict__ cb, float* __restrict__ xc)
{
  int idx = blockIdx.x * 256 + threadIdx.x;        // t*DINNER + d
  int d = idx & (DINNER - 1);
  int t = idx >> 11;
  int b = t >> 10;
  int l = t & (LL - 1);

  float acc = cb[d];
#pragma unroll
  for (int j = 0; j < 4; ++j) {
    int ll = l - 3 + j;
    if (ll >= 0)
      acc += cw[d * 4 + j] * xz[(size_t)(b * LL + ll) * 4096 + d];
  }
  xc[(size_t)t * DINNER + d] = acc / (1.f + __expf(-acc));   // SiLU
}

// ---------------------------------------------------------------------------
// selective scan + gating.  One thread per (b,d); h[16] in registers.
// B_t/C_t (shared across d) staged 16 steps at a time via LDS.
// Writes yact = (y + D*xc)*silu(z) over the dead xc half of xz.
// ---------------------------------------------------------------------------
__global__ __launch_bounds__(256) void scan_gate(
    const float* __restrict__ xc, const float* __restrict__ dtb,
    const float* __restrict__ xdbl, float* __restrict__ xz,
    const float* __restrict__ A_log, const float* __restrict__ Dp)
{
  __shared__ float sBC[16 * 32];   // 16 steps x (B[16] | C[16])
  const int b = blockIdx.y;
  const int d = blockIdx.x * 256 + threadIdx.x;

  float An[NSTATE], h[NSTATE];
#pragma unroll
  for (int n = 0; n < NSTATE; ++n) {
    An[n] = -__expf(A_log[(size_t)d * NSTATE + n]);
    h[n]  = 0.f;
  }
  const float Dd = Dp[d];

  for (int lc = 0; lc < LL / 16; ++lc) {
#pragma unroll
    for (int it = 0; it < 2; ++it) {           // stage 16 steps of B|C
      int i = threadIdx.x + it * 256;
      int s = i >> 5, n = i & 31;
      int t = b * LL + lc * 16 + s;
      sBC[s * 32 + n] = xdbl[(size_t)t * XDBL + RRANK + n];
    }
    __syncthreads();
    for (int s = 0; s < 16; ++s) {
      int t = b * LL + lc * 16 + s;
      float dtv = dtb[(size_t)t * DINNER + d];
      float xv  = xc [(size_t)t * DINNER + d];
      float sx  = dtv * xv;
      float y   = 0.f;
#pragma unroll
      for (int n = 0; n < NSTATE; ++n) {
        h[n] = __expf(dtv * An[n]) * h[n] + sx * sBC[s * 32 + n];
        y   += h[n] * sBC[s * 32 + 16 + n];
      }
      float zv = xz[(size_t)t * 4096 + DINNER + d];
      float g  = zv / (1.f + __expf(-zv));                 // silu(z)
      xz[(size_t)t * 4096 + d] = (y + Dd * xv) * g;        // gated output
    }
    __syncthreads();
  }
}

// ---------------------------------------------------------------------------
extern "C" void kernel_launch(void* const* d_in, const int* in_sizes, int n_in,
                              void* d_out, int out_size, void* d_ws, size_t ws_size,
                              hipStream_t stream)
{
  const float* y        = (const float*)d_in[0];
  const float* W_in_l   = (const float*)d_in[1];
  const float* b_in_l   = (const float*)d_in[2];
  const float* W_inproj = (const float*)d_in[3];
  const float* conv_w   = (const float*)d_in[4];
  const float* conv_b   = (const float*)d_in[5];
  const float* W_xproj  = (const float*)d_in[6];
  const float* W_dt     = (const float*)d_in[7];
  const float* b_dt     = (const float*)d_in[8];
  const float* A_log    = (const float*)d_in[9];
  const float* Dp       = (const float*)d_in[10];
  const float* W_outproj= (const float*)d_in[11];
  const float* W_out_l  = (const float*)d_in[12];
  const float* b_out_l  = (const float*)d_in[13];

  float* ws       = (float*)d_ws;
  float* buf_x    = ws;                                    // (T, DM)
  float* buf_x2   = buf_x    + (size_t)T_TOK * DMODEL;     // (T, DM)
  float* buf_xz   = buf_x2   + (size_t)T_TOK * DMODEL;     // (T, 4096)
  float* buf_xc   = buf_xz   + (size_t)T_TOK * 4096;       // (T, DI)
  float* buf_xdbl = buf_xc   + (size_t)T_TOK * DINNER;     // (T, 96)
  float* buf_dt   = buf_xdbl + (size_t)T_TOK * XDBL;       // (T, DI)

  const dim3 blk(256);
  const int GY = T_TOK / 128;

  // x = y @ W_in_l^T + b_in_l        (4096 x 1024 x 128)
  gemm_wmma_f32<<<dim3(DMODEL / 128, GY), blk, 0, stream>>>(
      y, OBSD, W_in_l, OBSD, b_in_l, buf_x, DMODEL, DMODEL, OBSD, 0);

  const float* xin = buf_x;
  for (int i = 0; i < NLAY; ++i) {
    const float* Wip = W_inproj  + (size_t)i * 2 * DINNER * DMODEL;
    const float* cw  = conv_w    + (size_t)i * DINNER * 4;
    const float* cb  = conv_b    + (size_t)i * DINNER;
    const float* Wxp = W_xproj   + (size_t)i * XDBL * DINNER;
    const float* Wdt = W_dt      + (size_t)i * DINNER * RRANK;
    const float* bdt = b_dt      + (size_t)i * DINNER;
    const float* Al  = A_log     + (size_t)i * DINNER * NSTATE;
    const float* Di  = Dp        + (size_t)i * DINNER;
    const float* Wop = W_outproj + (size_t)i * DMODEL * DINNER;
    float* xout = (i & 1) ? buf_x : buf_x2;

    // xz = x @ W_inproj^T            (4096 x 4096 x 1024)
    gemm_wmma_f32<<<dim3(4096 / 128, GY), blk, 0, stream>>>(
        xin, DMODEL, Wip, DMODEL, nullptr, buf_xz, 4096, 4096, DMODEL, 0);

    // depthwise causal conv + SiLU
    conv_silu<<<(T_TOK * DINNER) / 256, blk, 0, stream>>>(buf_xz, cw, cb, buf_xc);

    // x_dbl = xc @ W_xproj^T         (4096 x 96 x 2048)
    gemm_wmma_f32<<<dim3(1, GY), blk, 0, stream>>>(
        buf_xc, DINNER, Wxp, DINNER, nullptr, buf_xdbl, XDBL, XDBL, DINNER, 0);

    // dt = softplus(x_dbl[:, :64] @ W_dt^T + b_dt)   (4096 x 2048 x 64)
    gemm_wmma_f32<<<dim3(DINNER / 128, GY), blk, 0, stream>>>(
        buf_xdbl, XDBL, Wdt, RRANK, bdt, buf_dt, DINNER, DINNER, RRANK, 1);

    // selective scan + gating (writes yact into xc-half of xz)
    scan_gate<<<dim3(DINNER / 256, BB), blk, 0, stream>>>(
        buf_xc, buf_dt, buf_xdbl, buf_xz, Al, Di);

    // x = yact @ W_outproj^T         (4096 x 1024 x 2048)
    gemm_wmma_f32<<<dim3(DMODEL / 128, GY), blk, 0, stream>>>(
        buf_xz, 4096, Wop, DINNER, nullptr, xout, DMODEL, DMODEL, DINNER, 0);

    xin = xout;
  }

  // out = x @ W_out_l^T + b_out_l    (4096 x 32 x 1024)
  gemm_wmma_f32<<<dim3(1, GY), blk, 0, stream>>>(
      xin, DMODEL, W_out_l, DMODEL, b_out_l, (float*)d_out, ACTD, ACTD, DMODEL, 0);
}